// CellsortHamiltonian_60215441490504
// MI455X (gfx1250) — compile-verified
//
#include <hip/hip_runtime.h>
#include <hip/hip_bf16.h>

#define HH 4096
#define WW 4096
#define NIDS 200

typedef __attribute__((ext_vector_type(2))) float v2f;
typedef __attribute__((ext_vector_type(8))) float v8f;

__device__ __forceinline__ float softplusf(float x) {
    return (x > 20.0f) ? x : log1pf(expf(x));
}

// ---------------- kernel 0: zero global histogram ----------------
__global__ void cpm_zero(unsigned* __restrict__ ghist) {
    int t = threadIdx.x;
    if (t < NIDS) ghist[t] = 0u;
}

// ---------------- kernel 1: per-row interaction + histogram ----------------
// Block = one row. Direct coalesced global loads; the 128MB dataset fits the
// 192MB L2, so the "down" row re-read is an L2 hit. Integer atomics only
// (deterministic); float interaction energy goes to per-block partials.
__global__ __launch_bounds__(256) void cpm_main(
    const int* __restrict__ ids, const int* __restrict__ typ,
    const float* __restrict__ J, const float* __restrict__ gJ,
    const float* __restrict__ bJ, unsigned* __restrict__ ghist,
    float* __restrict__ partials)
{
    __shared__ float    Jeff[12];
    __shared__ unsigned lhist[256];
    __shared__ float    red[256];

    int tid = threadIdx.x;
    lhist[tid] = 0u;
    if (tid < 9) Jeff[tid] = softplusf(gJ[0]) * J[tid] + bJ[0];
    __syncthreads();

    int i  = blockIdx.x;
    int i2 = (i + 1) & (HH - 1);
    size_t rc = (size_t)i  * WW;
    size_t rd = (size_t)i2 * WW;
    const int4* idsC4 = (const int4*)(ids + rc);
    const int4* idsD4 = (const int4*)(ids + rd);
    const int4* typC4 = (const int4*)(typ + rc);
    const int4* typD4 = (const int4*)(typ + rd);

    float e = 0.0f;
    for (int c4 = tid; c4 < WW / 4; c4 += 256) {
        int j = c4 * 4;
        int4 ic  = idsC4[c4];
        int4 idn = idsD4[c4];
        int4 tc  = typC4[c4];
        int4 td  = typD4[c4];
        int jr = (j + 4) & (WW - 1);
        int jl = (j - 1) & (WW - 1);
        int icr = ids[rc + jr];
        int tcr = typ[rc + jr];
        int idr = ids[rd + jr];
        int tdr = typ[rd + jr];
        int idl = ids[rd + jl];
        int tdl = typ[rd + jl];

        atomicAdd(&lhist[ic.x], 1u);
        atomicAdd(&lhist[ic.y], 1u);
        atomicAdd(&lhist[ic.z], 1u);
        atomicAdd(&lhist[ic.w], 1u);

        int cc[5] = {ic.x, ic.y, ic.z, ic.w, icr};
        int ct[5] = {tc.x, tc.y, tc.z, tc.w, tcr};
        int dd[6] = {idl, idn.x, idn.y, idn.z, idn.w, idr};
        int dt[6] = {tdl, td.x, td.y, td.z, td.w, tdr};

        #pragma unroll
        for (int p = 0; p < 4; ++p) {
            int cid  = cc[p];
            int base = ct[p] * 3;
            float jR  = Jeff[base + ct[p + 1]];   // offset (0, 1)
            float jD  = Jeff[base + dt[p + 1]];   // offset (1, 0)
            float jDR = Jeff[base + dt[p + 2]];   // offset (1, 1)
            float jDL = Jeff[base + dt[p]];       // offset (1,-1)
            e += (cid != cc[p + 1]) ? jR  : 0.0f;
            e += (cid != dd[p + 1]) ? jD  : 0.0f;
            e += (cid != dd[p + 2]) ? jDR : 0.0f;
            e += (cid != dd[p])     ? jDL : 0.0f;
        }
    }

    __syncthreads();
    red[tid] = e;
    __syncthreads();
    #pragma unroll
    for (int s = 128; s > 0; s >>= 1) {
        if (tid < s) red[tid] += red[tid + s];
        __syncthreads();
    }
    if (tid == 0) partials[blockIdx.x] = red[0];
    if (tid < NIDS) {
        unsigned v = lhist[tid];
        if (v) atomicAdd(&ghist[tid], v);   // integer: order-independent
    }
}

// ---------------- kernel 2: finalize (1 wave, WMMA-based reduction) ------
// Uses V_WMMA_F32_16X16X4_F32 as a "sum 64 f32" primitive: A = data in the
// documented 16x4 layout, B = all-ones (layout independent), C accumulates.
// Partials are staged to LDS via gfx1250 async global->LDS copies.
__global__ __launch_bounds__(32) void cpm_final(
    const unsigned* __restrict__ ghist, const float* __restrict__ partials,
    const float* __restrict__ vp, const float* __restrict__ lam,
    const float* __restrict__ off, const float* __restrict__ offsc,
    float* __restrict__ out)
{
    __shared__ float sp_buf[4096];   // staged block partials
    __shared__ float sd[256];        // squared volume deviations (scaled)
    __shared__ float red2[32];

    int lane = threadIdx.x;

    // Async-stage 4096 partials (16B per lane-op) into LDS.
    for (int k = lane; k < 1024; k += 32) {
        unsigned ldsoff = (unsigned)(size_t)&sp_buf[k * 4];
        unsigned long long gaddr = (unsigned long long)(const void*)(partials + k * 4);
        asm volatile("global_load_async_to_lds_b128 %0, %1, off"
                     :: "v"(ldsoff), "v"(gaddr) : "memory");
    }
    asm volatile("s_wait_asynccnt 0" ::: "memory");
    __syncthreads();

    float vpref    = vp[0];
    float volScale = softplusf(lam[0]) + 0.001f;
    #pragma unroll
    for (int t = 0; t < 8; ++t) {
        int v = lane + 32 * t;
        float dv = 0.0f;
        if (v >= 1 && v < NIDS) {
            float d = (float)ghist[v] - vpref;
            dv = d * d * volScale;
        }
        sd[v] = dv;
    }
    __syncthreads();

    // A 16x4 f32 layout: VGPR0 = K(0|2), VGPR1 = K(1|3); lanes 0-15: M, K=0/1;
    // lanes 16-31: M, K=2/3.  Value index v = M*4 + K.
    int m  = lane & 15;
    int h  = lane >> 4;
    int i0 = m * 4 + h * 2;

    v8f c = {};
    v2f b; b.x = 1.0f; b.y = 1.0f;

    for (int it = 0; it < 64; ++it) {          // interaction partials (x 1/4)
        v2f a;
        a.x = sp_buf[it * 64 + i0]     * 0.25f;
        a.y = sp_buf[it * 64 + i0 + 1] * 0.25f;
        c = __builtin_amdgcn_wmma_f32_16x16x4_f32(false, a, false, b,
                                                  (short)0, c, false, false);
    }
    for (int it = 0; it < 4; ++it) {           // volume term
        v2f a;
        a.x = sd[it * 64 + i0];
        a.y = sd[it * 64 + i0 + 1];
        c = __builtin_amdgcn_wmma_f32_16x16x4_f32(false, a, false, b,
                                                  (short)0, c, false, false);
    }

    // D layout: VGPR e -> M=e (lanes 0-15, N=lane) / M=8+e (lanes 16-31).
    // All columns identical (B=1), so total = col0 rows 0-7 (lane0) + rows
    // 8-15 (lane16).
    float s = 0.0f;
    #pragma unroll
    for (int e = 0; e < 8; ++e) s += c[e];
    red2[lane] = s;
    __syncthreads();
    if (lane == 0) out[0] = red2[0] + red2[16] + off[0] * offsc[0];
}

extern "C" void kernel_launch(void* const* d_in, const int* in_sizes, int n_in,
                              void* d_out, int out_size, void* d_ws, size_t ws_size,
                              hipStream_t stream) {
    const int*   ids  = (const int*)d_in[0];
    const int*   typ  = (const int*)d_in[1];
    const float* J    = (const float*)d_in[2];
    const float* gJ   = (const float*)d_in[3];
    const float* bJ   = (const float*)d_in[4];
    const float* vp   = (const float*)d_in[5];
    const float* lam  = (const float*)d_in[6];
    const float* off  = (const float*)d_in[7];
    const float* offs = (const float*)d_in[8];

    unsigned* ghist    = (unsigned*)d_ws;            // 200 bins (pad to 256)
    float*    partials = (float*)d_ws + 256;         // 4096 block partials

    cpm_zero <<<1, 256, 0, stream>>>(ghist);
    cpm_main <<<HH, 256, 0, stream>>>(ids, typ, J, gJ, bJ, ghist, partials);
    cpm_final<<<1, 32, 0, stream>>>(ghist, partials, vp, lam, off, offs,
                                    (float*)d_out);
}